// _DirectionalStateScan2D_42107859370354
// MI455X (gfx1250) — compile-verified
//
#include <hip/hip_runtime.h>

// DirectionalStateScan2D on MI455X (gfx1250, wave32, WMMA).
//
// Plan: 1 workgroup (256 thr = 8 waves) per (b,c) image of 128x128 f32.
//   - Stage image once in LDS (~66 KiB padded), accumulate output in a
//     second LDS tile -> total HBM traffic = read once + write once (optimal;
//     memory-bound problem, ~3 flop/byte, floor ~11.5us at 23.3 TB/s).
//   - Each directional scan is a blocked matmul-scan with
//     V_WMMA_F32_16X16X4_F32: Y[16x16] = A[16x16] * X[16x16] + carry,
//     A lower-triangular of decay powers (K=16 via 4 chained K=4 WMMAs).
//   - Carry between 16-wide segments enters through the WMMA C operand and
//     is extracted from D (M=15 row -> VGPR7 of hi lanes) via lane shuffle.
//   - Lane-varying decay powers computed with v_exp/v_log (no dynamically
//     indexed private arrays -> no v_cndmask select trees).

#define HW   128
#define CCH  256
#define LDSP 132   // padded LDS row stride (floats): 16B-aligned rows, no bank conflicts

typedef float v2f __attribute__((ext_vector_type(2)));
typedef float v8f __attribute__((ext_vector_type(8)));

template <bool FIRST, bool VERT, bool FLIP>
__device__ __forceinline__ void dir_scan(const float* __restrict__ xin,
                                         float* __restrict__ acc,
                                         float d, float scale, float mix,
                                         int base, int tM, int hi)
{
    const float l2d = __log2f(d);            // v_log_f32
    const float s1d = scale * (1.0f - d);

    // A-matrix fragments (16x4 f32 layout: lanes 0-15 hold K=4q+{0,1},
    // lanes 16-31 hold K=4q+{2,3}; M = lane&15). Constant across segments.
    // A[t,j] = (j<=t) ? scale*(1-d)*d^(t-j) : 0, via d^e = exp2(e*log2 d).
    v2f a[4];
    #pragma unroll
    for (int q = 0; q < 4; ++q) {
        const int e0 = tM - (4 * q + 2 * hi);   // exponent for a.x
        const int e1 = e0 - 1;                  // exponent for a.y
        a[q].x = (e0 >= 0) ? s1d * __builtin_amdgcn_exp2f((float)e0 * l2d) : 0.0f;
        a[q].y = (e1 >= 0) ? s1d * __builtin_amdgcn_exp2f((float)e1 * l2d) : 0.0f;
    }

    // carry coefficients d^(M+1), M = k + 8*hi: exact multiply chain,
    // constant-indexed (no dynamic private-array indexing).
    const float d2 = d * d, d4 = d2 * d2, d8 = d4 * d4;
    const float dbase = hi ? (d8 * d) : d;      // d^(8*hi + 1)
    float dcar[8];
    dcar[0] = dbase;
    #pragma unroll
    for (int k = 1; k < 8; ++k) dcar[k] = dcar[k - 1] * d;

    float prev = 0.0f;                       // per-line carry (indexed by N)
    for (int s = 0; s < 8; ++s) {            // 8 serial 16-step segments
        v8f yv;
        #pragma unroll
        for (int k = 0; k < 8; ++k) yv[k] = dcar[k] * prev;   // C = carry term

        #pragma unroll
        for (int q = 0; q < 4; ++q) {        // chained K=4 WMMAs -> K=16
            const int k0 = 4 * q + 2 * hi;
            const int t0 = s * 16 + k0;
            v2f b;
            if (!VERT) {
                const int row = base + tM;
                const int c0 = FLIP ? (127 - t0)     : t0;
                const int c1 = FLIP ? (127 - t0 - 1) : (t0 + 1);
                b.x = xin[row * LDSP + c0];
                b.y = xin[row * LDSP + c1];
            } else {
                const int col = base + tM;
                const int r0 = FLIP ? (127 - t0)     : t0;
                const int r1 = FLIP ? (127 - t0 - 1) : (t0 + 1);
                b.x = xin[r0 * LDSP + col];
                b.y = xin[r1 * LDSP + col];
            }
            // D = A*B + C   (f32, full precision)
            yv = __builtin_amdgcn_wmma_f32_16x16x4_f32(
                     false, a[q], false, b, (short)0, yv, false, false);
        }

        // next-segment carry: y at M=15 lives in VGPR7 of lanes 16-31, N=lane&15
        prev = __shfl(yv[7], 16 + tM, 32);

        // accumulate mixed result into output tile
        #pragma unroll
        for (int k = 0; k < 8; ++k) {
            const int t = s * 16 + k + 8 * hi;
            int idx;
            if (!VERT) {
                const int row = base + tM;
                const int col = FLIP ? (127 - t) : t;
                idx = row * LDSP + col;
            } else {
                const int col = base + tM;
                const int row = FLIP ? (127 - t) : t;
                idx = row * LDSP + col;
            }
            if (FIRST) acc[idx]  = mix * yv[k];
            else       acc[idx] += mix * yv[k];
        }
    }
}

__global__ __launch_bounds__(256, 2)
void dss2d_kernel(const float* __restrict__ x,
                  const float* __restrict__ decay_logits,
                  const float* __restrict__ mix_logits,
                  const float* __restrict__ input_scale,
                  float* __restrict__ out)
{
    extern __shared__ float smem[];
    float* xin = smem;               // [128][LDSP] staged input
    float* acc = smem + HW * LDSP;   // [128][LDSP] mixed output accumulator

    const int bc   = blockIdx.x;     // b*C + c
    const int c    = bc & (CCH - 1);
    const int tid  = threadIdx.x;    // 0..255 (8 wave32)
    const int lane = tid & 31;
    const int wv   = tid >> 5;       // 0..7 -> owns 16 lines
    const int tM   = lane & 15;
    const int hi   = lane >> 4;
    const int base = 16 * wv;

    // ---- per-channel parameters (tiny; computed redundantly per thread) ----
    float dly[4], mx[4], sc[4];
    {
        float ex[4], esum = 0.0f;
        #pragma unroll
        for (int k = 0; k < 4; ++k) {
            const float dl = decay_logits[k * CCH + c];
            float d = 1.0f / (1.0f + __expf(-dl));
            dly[k] = fminf(fmaxf(d, 0.05f), 0.995f);
            ex[k] = __expf(mix_logits[k * CCH + c]);
            esum += ex[k];
            sc[k] = 1.0f + tanhf(input_scale[k * CCH + c]);
        }
        const float r = 1.0f / esum;
        #pragma unroll
        for (int k = 0; k < 4; ++k) mx[k] = ex[k] * r;
    }

    // ---- stage image: global -> LDS, coalesced 16B loads ----
    const float* xg = x + (size_t)bc * (HW * HW);
    for (int e4 = tid; e4 < (HW * HW) / 4; e4 += 256) {
        const int fe  = e4 * 4;
        const int row = fe >> 7;
        const int col = fe & 127;
        const float4 v = ((const float4*)xg)[e4];
        *(float4*)&xin[row * LDSP + col] = v;
    }
    __syncthreads();

    // ---- four directional WMMA matmul-scans ----
    dir_scan<true,  false, false>(xin, acc, dly[0], sc[0], mx[0], base, tM, hi); // L->R
    __syncthreads();
    dir_scan<false, false, true >(xin, acc, dly[1], sc[1], mx[1], base, tM, hi); // R->L
    __syncthreads();
    dir_scan<false, true,  false>(xin, acc, dly[2], sc[2], mx[2], base, tM, hi); // T->B
    __syncthreads();
    dir_scan<false, true,  true >(xin, acc, dly[3], sc[3], mx[3], base, tM, hi); // B->T
    __syncthreads();

    // ---- writeback: LDS -> global, coalesced 16B stores ----
    float* og = out + (size_t)bc * (HW * HW);
    for (int e4 = tid; e4 < (HW * HW) / 4; e4 += 256) {
        const int fe  = e4 * 4;
        const int row = fe >> 7;
        const int col = fe & 127;
        ((float4*)og)[e4] = *(const float4*)&acc[row * LDSP + col];
    }
}

extern "C" void kernel_launch(void* const* d_in, const int* in_sizes, int n_in,
                              void* d_out, int out_size, void* d_ws, size_t ws_size,
                              hipStream_t stream)
{
    const float* x  = (const float*)d_in[0];
    const float* dl = (const float*)d_in[1];
    const float* ml = (const float*)d_in[2];
    const float* is = (const float*)d_in[3];
    float* out = (float*)d_out;

    const int nbc = in_sizes[0] / (HW * HW);              // B*C = 2048
    const size_t shmem = (size_t)2 * HW * LDSP * sizeof(float); // ~132 KiB of 320 KiB LDS

    dss2d_kernel<<<dim3(nbc), dim3(256), shmem, stream>>>(x, dl, ml, is, out);
}